// KalmanFilter_83726092468659
// MI455X (gfx1250) — compile-verified
//
#include <hip/hip_runtime.h>

typedef __attribute__((ext_vector_type(2))) float v2f;
typedef __attribute__((ext_vector_type(8))) float v8f;

// F with dt = 1:  F[i][j] = 1 if j==i, j==i+2, or j==i+4
__device__ __forceinline__ float Fmat(int i, int j) {
    return (j == i || j == i + 2 || j == i + 4) ? 1.0f : 0.0f;
}

// ---------------------------------------------------------------------------
// Phase 1: serial covariance recursion (batch-independent).
// Emits the WMMA A-operand stream for V_WMMA_F32_16X16X4_F32, pre-swizzled
// per lane. Abig (16x8) = [ A_t (6x6) | K_t (6x2) ] with:
//   rows 0..5  = [A|K]
//   rows 8..11 = rows 2..5   (duplication trick: makes the WMMA result carry
//                             X rows 2..5 in the upper half-wave's acc[0..3],
//                             so next step's B operand needs no shuffles)
//   rows 6,7, 12..15 = 0
// Per-lane float4 (m = lane&15):
//   lanes 0-15 : { Abig[m][0], Abig[m][1], Abig[m][4], Abig[m][5] }
//   lanes 16-31: { Abig[m][2], Abig[m][3], Abig[m][6], Abig[m][7] }
// ---------------------------------------------------------------------------
__global__ void kf_gain_kernel(const float* __restrict__ Qmul,
                               const float* __restrict__ R,
                               float* __restrict__ aPacked, int T) {
    __shared__ float sP[36], sFP[36], sPp[36], sK[12], sA[36];
    const int tid = threadIdx.x;
    const int i = tid / 6, j = tid % 6;
    const float qm  = Qmul[0];
    const float R00 = R[0], R01 = R[1], R10 = R[2], R11 = R[3];

    float Qij = 0.0f;
    if (tid < 36) {
#pragma unroll
        for (int k = 0; k < 6; ++k) Qij += Fmat(i, k) * Fmat(j, k);
        Qij *= qm;                              // Q = (F F^T) * q_mul
        sP[tid] = (i == j) ? 100.0f : 0.0f;     // P0 = 100 * I
    }
    __syncthreads();

    for (int t = 0; t < T; ++t) {
        if (tid < 36) {                         // FP = F @ P
            float s = 0.0f;
#pragma unroll
            for (int k = 0; k < 6; ++k) s += Fmat(i, k) * sP[k * 6 + j];
            sFP[tid] = s;
        }
        __syncthreads();
        if (tid < 36) {                         // Pp = FP @ F^T + Q
            float s = Qij;
#pragma unroll
            for (int k = 0; k < 6; ++k) s += sFP[i * 6 + k] * Fmat(j, k);
            sPp[tid] = s;
        }
        __syncthreads();
        {   // S = H Pp H^T + R (2x2), K = Pp H^T S^-1
            float S00 = sPp[0] + R00, S01 = sPp[1] + R01;
            float S10 = sPp[6] + R10, S11 = sPp[7] + R11;
            float inv = 1.0f / (S00 * S11 - S01 * S10);
            float i00 =  S11 * inv, i01 = -S01 * inv;
            float i10 = -S10 * inv, i11 =  S00 * inv;
            if (tid < 12) {
                int ki = tid >> 1, kj = tid & 1;
                sK[tid] = sPp[ki * 6 + 0] * (kj ? i01 : i00)
                        + sPp[ki * 6 + 1] * (kj ? i11 : i10);
            }
        }
        __syncthreads();
        if (tid < 36) {                         // Pn = (I-KH)Pp ; A = (I-KH)F
            float k0 = sK[i * 2 + 0], k1 = sK[i * 2 + 1];
            float Pn = sPp[tid] - k0 * sPp[j] - k1 * sPp[6 + j];
            sA[tid]  = Fmat(i, j) - k0 * Fmat(0, j) - k1 * Fmat(1, j);
            sP[tid]  = Pn;
        }
        __syncthreads();
        if (tid < 32) {                         // pack WMMA A-operand stream
            const int  m  = tid & 15;
            const bool lo = tid < 16;
            // effective source row of Abig for matrix row m
            const int  r  = (m < 6) ? m : ((m >= 8 && m <= 11) ? (m - 6) : -1);
            float v0 = 0.f, v1 = 0.f, v2 = 0.f, v3 = 0.f;
            if (r >= 0) {
                auto AB = [&](int c) -> float {
                    return (c < 6) ? sA[r * 6 + c] : sK[r * 2 + (c - 6)];
                };
                v0 = AB(lo ? 0 : 2);  v1 = AB(lo ? 1 : 3);
                v2 = AB(lo ? 4 : 6);  v3 = AB(lo ? 5 : 7);
            }
            reinterpret_cast<float4*>(aPacked)[t * 32 + tid] =
                make_float4(v0, v1, v2, v3);
        }
        __syncthreads();
    }
}

// ---------------------------------------------------------------------------
// Phase 2: batched affine scan  X_t = A_t X_{t-1} + K_t z_t  via f32 WMMA.
// One wave32 owns 16 batch columns. Row-duplicated Abig means:
//   lanes 0-15 : acc[r] = X row r       (r < 6)
//   lanes 16-31: acc[r] = X row r+2     (r < 4), acc[4..7] = 0
// so B0 = {acc[0],acc[1]} in every lane (rows 0/1 low, 2/3 high) and only
// the z injection (rows 6,7, upper half) needs cndmasks. No shuffles, no
// exec divergence in the hot loop.
// ---------------------------------------------------------------------------
__global__ void __launch_bounds__(256)
kf_scan_kernel(const float* __restrict__ meas, const float* __restrict__ tpos,
               const float* __restrict__ X0, const float* __restrict__ aPacked,
               float* __restrict__ out, int T, int nTiles, float scale) {
    const int  lane = threadIdx.x & 31;
    const int  wave = threadIdx.x >> 5;
    const int  tile = blockIdx.x * (blockDim.x >> 5) + wave;
    if (tile >= nTiles) return;                 // uniform per wave
    const int  col  = lane & 15;
    const bool lo   = lane < 16;
    const float msk = lo ? 1.0f : 0.0f;
    const long long b = (long long)tile * 16 + col;

    // Initial state in the layout the WMMA would have produced.
    const float* x0 = X0 + b * 6;               // both halves read same column
    v8f acc = {};
    if (lo) {
        acc[0] = x0[0]; acc[1] = x0[1]; acc[2] = x0[2];
        acc[3] = x0[3]; acc[4] = x0[4]; acc[5] = x0[5];
    } else {
        acc[0] = x0[2]; acc[1] = x0[3]; acc[2] = x0[4]; acc[3] = x0[5];
    }

    const float2* zp  = reinterpret_cast<const float2*>(meas) + b * T;
    const float2* tpp = reinterpret_cast<const float2*>(tpos) + b * T;
    const float2* mp  = lo ? tpp : zp;          // lower: truth, upper: z
    const float4* ap  = reinterpret_cast<const float4*>(aPacked);

    float  sumsq = 0.0f;
    float4 a = ap[lane];                        // t = 0 operands (prefetched)
    float2 m = mp[0];

    for (int t = 0; t < T; ++t) {
        const int tn = (t + 1 < T) ? (t + 1) : t;
        float4 a_n = ap[tn * 32 + lane];        // prefetch next step
        float2 m_n = mp[tn];

        v2f A0 = { a.x, a.y };
        v2f A1 = { a.z, a.w };
        v2f B0 = { acc[0], acc[1] };            // rows 0/1 (low), 2/3 (high)
        v2f B1 = { lo ? acc[4] : m.x,           // rows 4/5 (low), z (high)
                   lo ? acc[5] : m.y };

        v8f czero = {};
        v8f d1 = __builtin_amdgcn_wmma_f32_16x16x4_f32(
                     false, A1, false, B1, (short)0, czero, false, false);
        acc    = __builtin_amdgcn_wmma_f32_16x16x4_f32(
                     false, A0, false, B0, (short)0, d1, false, false);

        // pred = (F Xn)[:2] with dt = 1; only lanes 0-15 contribute (mask)
        float px = acc[0] + acc[2] + acc[4];
        float py = acc[1] + acc[3] + acc[5];
        float ex = px - m.x, ey = py - m.y;     // m = truth on lower half
        float e2 = fmaf(ex, ex, ey * ey);
        sumsq = fmaf(e2, msk, sumsq);

        a = a_n; m = m_n;
    }

#pragma unroll
    for (int off = 16; off > 0; off >>= 1)
        sumsq += __shfl_xor(sumsq, off, 32);
    if (lane == 0) atomicAdd(out, sumsq * scale);
}

extern "C" void kernel_launch(void* const* d_in, const int* in_sizes, int n_in,
                              void* d_out, int out_size, void* d_ws, size_t ws_size,
                              hipStream_t stream) {
    const float* meas = (const float*)d_in[0];
    const float* tpos = (const float*)d_in[1];
    const float* X0   = (const float*)d_in[2];
    const float* Qm   = (const float*)d_in[3];
    const float* R    = (const float*)d_in[4];
    float* out = (float*)d_out;

    const int B = in_sizes[2] / 6;
    const int T = in_sizes[0] / (B * 2);
    float* aPacked = (float*)d_ws;              // needs T*128 floats (512 KB)

    hipMemsetAsync(out, 0, sizeof(float), stream);
    kf_gain_kernel<<<1, 64, 0, stream>>>(Qm, R, aPacked, T);

    const int nTiles = B / 16;                  // 512 waves total
    const int wavesPerBlock = 8;
    dim3 block(32 * wavesPerBlock);
    dim3 grid((nTiles + wavesPerBlock - 1) / wavesPerBlock);
    const float scale = 1.0f / ((float)B * (float)T * 2.0f);
    kf_scan_kernel<<<grid, block, 0, stream>>>(meas, tpos, X0, aPacked, out,
                                               T, nTiles, scale);
}